// LetterGNN_88252987998746
// MI455X (gfx1250) — compile-verified
//
#include <hip/hip_runtime.h>

#define HID 64

typedef float v2f __attribute__((ext_vector_type(2)));
typedef float v8f __attribute__((ext_vector_type(8)));

// ---------------------------------------------------------------- utilities
__global__ void zero_f32(float* __restrict__ p, long long n) {
  long long i = (long long)blockIdx.x * blockDim.x + threadIdx.x;
  long long stride = (long long)gridDim.x * blockDim.x;
  for (; i < n; i += stride) p[i] = 0.0f;
}

// ---------------------------------------------------------------- degrees
__global__ void deg_accum(const int* __restrict__ dst, float* __restrict__ deg, int nE) {
  int e = blockIdx.x * blockDim.x + threadIdx.x;
  if (e < nE) atomicAdd(&deg[dst[e]], 1.0f);
}

__global__ void deg_finalize(float* __restrict__ dinv, int n) {
  int i = blockIdx.x * blockDim.x + threadIdx.x;
  if (i < n) dinv[i] = rsqrtf(dinv[i] + 1.0f);  // +1 self loop; deg >= 1
}

// ---------------------------------------------------------------- WMMA GEMM
// out[rows x 64] = A[rows x 64] @ W[64 x 64]   (+ bias, relu when fused)
// One wave computes a 16x64 tile with v_wmma_f32_16x16x4_f32.
__global__ void gemm64_wmma(const float* __restrict__ A, const float* __restrict__ W,
                            const float* __restrict__ bias, float* __restrict__ out,
                            int rows, int fuseBiasRelu) {
  __shared__ float w[HID * HID];
  for (int i = threadIdx.x; i < HID * HID; i += blockDim.x) w[i] = W[i];
  __syncthreads();

  const int lane = threadIdx.x & 31;
  const int wave = threadIdx.x >> 5;
  const int tile = blockIdx.x * (blockDim.x >> 5) + wave;
  if (tile * 16 >= rows) return;                 // wave-uniform exit, EXEC stays full

  const int rlo   = lane & 15;                   // row-in-tile (A) / col (B,C)
  const int khalf = (lane >> 4) << 1;            // K offset: lanes16-31 -> +2
  const float* arow = A + (long long)(tile * 16 + rlo) * HID;

  v8f acc[4] = {v8f{}, v8f{}, v8f{}, v8f{}};

  for (int k0 = 0; k0 < HID; k0 += 4) {
    const int ka = k0 + khalf;
    v2f a;
    a.x = arow[ka];
    a.y = arow[ka + 1];
#pragma unroll
    for (int nt = 0; nt < 4; ++nt) {
      v2f b;
      b.x = w[ka * HID + nt * 16 + rlo];
      b.y = w[(ka + 1) * HID + nt * 16 + rlo];
      acc[nt] = __builtin_amdgcn_wmma_f32_16x16x4_f32(
          false, a, false, b, (short)0, acc[nt], false, false);
    }
  }

  const int mhi = (lane >> 4) << 3;              // lanes16-31 hold rows 8..15
  float* obase = out + (long long)tile * 16 * HID;
#pragma unroll
  for (int nt = 0; nt < 4; ++nt) {
    const int col = nt * 16 + rlo;
    const float bv = fuseBiasRelu ? bias[col] : 0.0f;
#pragma unroll
    for (int v = 0; v < 8; ++v) {
      float val = acc[nt][v] + bv;
      if (fuseBiasRelu) val = fmaxf(val, 0.0f);
      obase[(long long)(mhi + v) * HID + col] = val;
    }
  }
}

// ---------------------------------------------------------------- edge scatter
// agg[dst] += xw[src] * dinv[src]*dinv[dst];  thread = (edge, 4 features)
__global__ void edge_scatter(const int* __restrict__ src, const int* __restrict__ dst,
                             const float* __restrict__ dinv, const float* __restrict__ xw,
                             float* __restrict__ agg, int nE) {
  long long idx = (long long)blockIdx.x * blockDim.x + threadIdx.x;
  if (idx >= (long long)nE * 16) return;
  const int e = (int)(idx >> 4);
  const int c = (int)(idx & 15) * 4;
  const int s = src[e], d = dst[e];
  const float wgt = dinv[s] * dinv[d];
  const float4 v = *(const float4*)(xw + (long long)s * HID + c);
  float* o = agg + (long long)d * HID + c;
  atomicAdd(o + 0, v.x * wgt);
  atomicAdd(o + 1, v.y * wgt);
  atomicAdd(o + 2, v.z * wgt);
  atomicAdd(o + 3, v.w * wgt);
}

// agg = relu(agg + xw * dinv^2 + b)   (in place)
__global__ void self_bias_relu(const float* __restrict__ xw, const float* __restrict__ dinv,
                               const float* __restrict__ b, float* __restrict__ agg, int n) {
  long long idx = (long long)blockIdx.x * blockDim.x + threadIdx.x;
  if (idx >= (long long)n * HID) return;
  const int node = (int)(idx >> 6);
  const int c = (int)(idx & 63);
  const float di = dinv[node];
  const float v = agg[idx] + xw[idx] * di * di + b[c];
  agg[idx] = fmaxf(v, 0.0f);
}

// ---------------------------------------------------------------- pooling
__global__ void pool_accum(const float* __restrict__ h, const int* __restrict__ batch,
                           float* __restrict__ sums, float* __restrict__ cnts, int n) {
  long long idx = (long long)blockIdx.x * blockDim.x + threadIdx.x;
  if (idx >= (long long)n * HID) return;
  const int node = (int)(idx >> 6);
  const int c = (int)(idx & 63);
  const int g = batch[node];
  atomicAdd(&sums[(long long)g * HID + c], h[idx]);
  if (c == 0) atomicAdd(&cnts[g], 1.0f);
}

__global__ void pool_div(const float* __restrict__ sums, const float* __restrict__ cnts,
                         float* __restrict__ pooled, int total) {
  int idx = blockIdx.x * blockDim.x + threadIdx.x;
  if (idx < total) pooled[idx] = sums[idx] / fmaxf(cnts[idx >> 6], 1.0f);
}

// ---------------------------------------------------------------- FC2 head
__global__ void fc2_kernel(const float* __restrict__ a, const float* __restrict__ W,
                           const float* __restrict__ b, float* __restrict__ out, int G) {
  int idx = blockIdx.x * blockDim.x + threadIdx.x;
  if (idx >= G * 26) return;
  const int g = idx / 26;
  const int j = idx % 26;
  float s = b[j];
  const float* ar = a + (long long)g * HID;
#pragma unroll
  for (int k = 0; k < HID; ++k) s += ar[k] * W[k * 26 + j];
  out[idx] = s;
}

// ---------------------------------------------------------------- launcher
extern "C" void kernel_launch(void* const* d_in, const int* in_sizes, int n_in,
                              void* d_out, int out_size, void* d_ws, size_t ws_size,
                              hipStream_t stream) {
  const float* x     = (const float*)d_in[0];
  const int*   ei    = (const int*)d_in[1];
  const int*   batch = (const int*)d_in[2];
  const float* W0 = (const float*)d_in[3];  const float* b0 = (const float*)d_in[4];
  const float* W1 = (const float*)d_in[5];  const float* b1 = (const float*)d_in[6];
  const float* W2 = (const float*)d_in[7];  const float* b2 = (const float*)d_in[8];
  const float* fcW  = (const float*)d_in[9];  const float* fcb  = (const float*)d_in[10];
  const float* fc2W = (const float*)d_in[11]; const float* fc2b = (const float*)d_in[12];
  float* out = (float*)d_out;

  const int N = in_sizes[0] / HID;      // 100000
  const int E = in_sizes[1] / 2;        // 1000000
  const int G = out_size / 26;          // 512
  const int* src = ei;
  const int* dstv = ei + E;

  // workspace carve-up (floats, 256-float aligned)
  float* ws = (float*)d_ws;
  size_t off = 0;
  float* dinv = ws + off;  off += ((size_t)N + 255) & ~(size_t)255;
  float* bufA = ws + off;  off += (size_t)N * HID;          // xw
  float* bufB = ws + off;  off += (size_t)N * HID;          // agg / h
  float* sums = ws + off;  off += (size_t)G * HID;
  float* cnts = ws + off;  off += ((size_t)G + 255) & ~(size_t)255;
  float* pooled = ws + off; off += (size_t)G * HID;
  float* fc1 = ws + off;

  const dim3 blk(256);

  // degrees -> dinv
  zero_f32<<<1024, blk, 0, stream>>>(dinv, N);
  deg_accum<<<(E + 255) / 256, blk, 0, stream>>>(dstv, dinv, E);
  deg_finalize<<<(N + 255) / 256, blk, 0, stream>>>(dinv, N);

  const int tilesN  = (N + 15) / 16;
  const int blocksN = (tilesN + 7) / 8;     // 8 waves / block
  const long long nh = (long long)N * HID;
  const unsigned scatterBlocks = (unsigned)(((long long)E * 16 + 255) / 256);
  const unsigned nhBlocks = (unsigned)((nh + 255) / 256);

  const float* hin = x;
  const float* Ws[3] = {W0, W1, W2};
  const float* bs[3] = {b0, b1, b2};
  for (int L = 0; L < 3; ++L) {
    gemm64_wmma<<<blocksN, blk, 0, stream>>>(hin, Ws[L], nullptr, bufA, N, 0);
    zero_f32<<<4096, blk, 0, stream>>>(bufB, nh);
    edge_scatter<<<scatterBlocks, blk, 0, stream>>>(src, dstv, dinv, bufA, bufB, E);
    self_bias_relu<<<nhBlocks, blk, 0, stream>>>(bufA, dinv, bs[L], bufB, N);
    hin = bufB;
  }

  // global mean pool
  zero_f32<<<64, blk, 0, stream>>>(sums, (long long)G * HID);
  zero_f32<<<2, blk, 0, stream>>>(cnts, G);
  pool_accum<<<nhBlocks, blk, 0, stream>>>(bufB, batch, sums, cnts, N);
  pool_div<<<(G * HID + 255) / 256, blk, 0, stream>>>(sums, cnts, pooled, G * HID);

  // FC1 (WMMA, bias+relu fused), FC2 (scalar, 26 cols)
  const int tilesG = (G + 15) / 16;
  gemm64_wmma<<<(tilesG + 7) / 8, blk, 0, stream>>>(pooled, fcW, fcb, fc1, G, 1);
  fc2_kernel<<<(G * 26 + 255) / 256, blk, 0, stream>>>(fc1, fc2W, fc2b, out, G);
}